// MultiHeadMixedAttention_43104291783120
// MI455X (gfx1250) — compile-verified
//
#include <hip/hip_runtime.h>
#include <cstddef>

typedef __attribute__((ext_vector_type(2))) float v2f;
typedef __attribute__((ext_vector_type(8))) float v8f;
typedef __attribute__((ext_vector_type(4))) int v4i;

#define KK 8
#define RR 128

// workspace offsets in floats (total 3,145,728 floats = 12 MB)
#define OFF_XT 0u
#define OFF_YT 262144u
#define OFF_XV 524288u
#define OFF_U  786432u
#define OFF_V  1310720u
#define OFF_R  1835008u
#define OFF_T  2883584u

// padded LDS row strides (floats): keep 16B alignment, skew banks (stride % 64 != 0)
#define LD64  68
#define LD128 132

// ---- CDNA5 async global->LDS path ------------------------------------------
#if __has_builtin(__builtin_amdgcn_global_load_async_to_lds_b128)
#define HAS_ASYNC_LDS 1
#else
#define HAS_ASYNC_LDS 0
#endif

__device__ __forceinline__ void async_wait0() {
#if HAS_ASYNC_LDS
#if __has_builtin(__builtin_amdgcn_s_wait_asynccnt)
  __builtin_amdgcn_s_wait_asynccnt(0);
#else
  asm volatile("s_wait_asynccnt 0x0" ::: "memory");
#endif
#endif
}

// ---- WMMA helpers (f32 16x16x4, wave32) ------------------------------------
__device__ __forceinline__ v8f wmma4(v2f a, v2f b, v8f c) {
  return __builtin_amdgcn_wmma_f32_16x16x4_f32(false, a, false, b, (short)0, c,
                                               false, false);
}

// 16-row x 4-K operand fragment. base points at (tile row 0, k0).
// A layout: lanes 0-15 -> rows 0-15 with K={0,1}; lanes 16-31 -> K={2,3}.
__device__ __forceinline__ v2f frag_ks(const float* base, int rowStride, int kStride) {
  int lane = threadIdx.x & 31;
  const float* p = base + (lane & 15) * rowStride + ((lane >> 4) * 2) * kStride;
  v2f f;
  f.x = p[0];
  f.y = p[kStride];
  return f;
}

// C/D layout: VGPR i holds row (i + 8*(lane>>4)), col (lane&15).
__device__ __forceinline__ void tile_store(float* c, int ldc, v8f acc) {
  int lane = threadIdx.x & 31;
  int half = lane >> 4, col = lane & 15;
#pragma unroll
  for (int i = 0; i < 8; ++i) c[(i + 8 * half) * ldc + col] = acc[i];
}

__device__ __forceinline__ v8f tile_load(const float* c, int ldc) {
  int lane = threadIdx.x & 31;
  int half = lane >> 4, col = lane & 15;
  v8f a;
#pragma unroll
  for (int i = 0; i < 8; ++i) a[i] = c[(i + 8 * half) * ldc + col];
  return a;
}

// cooperative row-tiled copy global -> LDS, b128 per lane, async when available
__device__ __forceinline__ void cp_rows(float* dst, int dstStride, const float* src,
                                        long srcStride, int rows, int rowFloats) {
  int f4PerRow = rowFloats >> 2;
  int total = rows * f4PerRow;
  for (int j = threadIdx.x; j < total; j += 256) {
    int row = j / f4PerRow, c4 = j - row * f4PerRow;
    const float4* s = (const float4*)(src + (size_t)row * (size_t)srcStride) + c4;
    float4* d = (float4*)(dst + row * dstStride) + c4;
#if HAS_ASYNC_LDS
    __builtin_amdgcn_global_load_async_to_lds_b128((v4i*)s, (v4i*)d, 0, 0);
#else
    *d = *s;
#endif
  }
}

// ---- Kernel 1: xt / yt / xv projections  [256x512]@[512x64] ----------------
__global__ void k_proj(const float* __restrict__ x_k, const float* __restrict__ y_q,
                       const float* __restrict__ Lam_x, const float* __restrict__ Lam_y,
                       const float* __restrict__ Theta_x, float* __restrict__ ws) {
  extern __shared__ float lds[];
  float* aLds = lds;              // 64 x LD64
  float* bLds = lds + 64 * LD64;  // 64 x 64 raw [p][d] (lanes walk d: conflict-free)
  int blk = blockIdx.x;
  int fam = blk >> 6; // 0=xt 1=yt 2=xv
  int rem = blk & 63;
  int b = rem >> 5;
  int k = (rem >> 2) & 7;
  int mb = rem & 3;
  const float* src = (fam == 1) ? y_q : x_k;                            // [B,256,512]
  const float* W = (fam == 0) ? Lam_x : ((fam == 1) ? Lam_y : Theta_x); // [K,512,64]
  float* dst = ws + (fam == 0 ? OFF_XT : (fam == 1 ? OFF_YT : OFF_XV));
  int w = threadIdx.x >> 5;
  v8f acc0 = {}, acc1 = {};
  for (int kc = 0; kc < 8; ++kc) {
    __syncthreads();
    cp_rows(aLds, LD64, src + ((size_t)(b * 256 + mb * 64)) * 512 + kc * 64, 512, 64, 64);
    cp_rows(bLds, 64, W + ((size_t)k * 512 + kc * 64) * 64, 64, 64, 64);
    async_wait0();
    __syncthreads();
#pragma unroll
    for (int j = 0; j < 2; ++j) {
      int ti = w + j * 8;
      int mt = ti >> 2, nt = ti & 3;
      v8f acc = j ? acc1 : acc0;
      for (int k0 = 0; k0 < 64; k0 += 4) {
        v2f a = frag_ks(aLds + mt * 16 * LD64 + k0, LD64, 1);
        v2f bb = frag_ks(bLds + k0 * 64 + nt * 16, 1, 64); // B[p][d] transposed access
        acc = wmma4(a, bb, acc);
      }
      if (j) acc1 = acc; else acc0 = acc;
    }
  }
#pragma unroll
  for (int j = 0; j < 2; ++j) {
    int ti = w + j * 8;
    int mt = ti >> 2, nt = ti & 3;
    tile_store(dst + ((size_t)((b * KK + k) * 256 + mb * 64 + mt * 16)) * 64 + nt * 16, 64,
               j ? acc1 : acc0);
  }
}

// ---- Kernel 2: u = xt@Lam_zy^T, v = yt@Lam_zx^T  [64x64]@[64x128] ----------
__global__ void k_uv(const float* __restrict__ ws_xt, const float* __restrict__ ws_yt,
                     const float* __restrict__ Lam_zy, const float* __restrict__ Lam_zx,
                     float* __restrict__ u, float* __restrict__ v) {
  extern __shared__ float lds[];
  float* aLds = lds;             // 64 x LD64
  float* bLds = lds + 64 * LD64; // 128 x LD64 raw [r][d]
  int blk = blockIdx.x;
  int which = blk >> 6;
  int rem = blk & 63;
  int b = rem >> 5, k = (rem >> 2) & 7, mb = rem & 3;
  const float* A = (which ? ws_yt : ws_xt) + ((size_t)((b * KK + k) * 256 + mb * 64)) * 64;
  const float* W = (which ? Lam_zx : Lam_zy) + (size_t)k * RR * 64;
  float* dst = (which ? v : u) + ((size_t)((b * KK + k) * 256 + mb * 64)) * RR;
  cp_rows(aLds, LD64, A, 64, 64, 64);
  cp_rows(bLds, LD64, W, 64, 128, 64);
  async_wait0();
  __syncthreads();
  int w = threadIdx.x >> 5;
#pragma unroll
  for (int j = 0; j < 4; ++j) {
    int ti = w * 4 + j;
    int mt = ti >> 3, rt = ti & 7;
    v8f acc = {};
    for (int d0 = 0; d0 < 64; d0 += 4) {
      v2f a = frag_ks(aLds + mt * 16 * LD64 + d0, LD64, 1);
      v2f bb = frag_ks(bLds + rt * 16 * LD64 + d0, LD64, 1);
      acc = wmma4(a, bb, acc);
    }
    tile_store(dst + (size_t)(mt * 16) * RR + rt * 16, RR, acc);
  }
}

// ---- Kernel 3: r[b,k,m,n] = Z_m @ u_m^T  (per (b,m): [256x128]@[128x16pad]) -
__global__ void k_r2(const float* __restrict__ z, const float* __restrict__ u,
                     float* __restrict__ rws) {
  extern __shared__ float lds[];
  float* zLds = lds;               // 256 x LD128 (whole Z_m slab)
  float* uLds = lds + 256 * LD128; // 16 x LD128, rows 8..15 zero
  int b = blockIdx.x >> 8;
  int m = blockIdx.x & 255;
  cp_rows(zLds, LD128, z + ((size_t)(b * 256 + m)) * 256 * 128, 128, 256, 128);
  cp_rows(uLds, LD128, u + (((size_t)(b * KK)) * 256 + m) * RR, (long)256 * RR, 8, 128);
  for (int j = threadIdx.x; j < 8 * LD128; j += 256) uLds[8 * LD128 + j] = 0.f;
  async_wait0();
  __syncthreads();
  int w = threadIdx.x >> 5;
  int lane = threadIdx.x & 31, half = lane >> 4, col = lane & 15;
#pragma unroll
  for (int j = 0; j < 2; ++j) {
    int nt = w * 2 + j;
    v8f acc = {};
    for (int k0 = 0; k0 < 128; k0 += 4) {
      v2f a = frag_ks(zLds + nt * 16 * LD128 + k0, LD128, 1);
      v2f bb = frag_ks(uLds + k0, LD128, 1);
      acc = wmma4(a, bb, acc);
    }
    if (col < 8) { // only K=8 real output columns
      float* dst = rws + (((size_t)(b * KK + col) * 256 + m) * 256) + nt * 16 + 8 * half;
#pragma unroll
      for (int i = 0; i < 8; ++i) dst[i] = acc[i];
    }
  }
}

// ---- Kernel 4: r[b,k,m,n] += Z_n @ v_n^T (per (b,n): [256x128]@[128x16pad]) -
__global__ void k_r3(const float* __restrict__ z, const float* __restrict__ v,
                     float* __restrict__ rws) {
  extern __shared__ float lds[];
  float* zLds = lds;               // 256 rows (m) x LD128
  float* vLds = lds + 256 * LD128; // 16 x LD128
  int b = blockIdx.x >> 8;
  int n = blockIdx.x & 255;
  cp_rows(zLds, LD128, z + (((size_t)(b * 256)) * 256 + n) * 128, (long)256 * 128, 256, 128);
  cp_rows(vLds, LD128, v + (((size_t)(b * KK)) * 256 + n) * RR, (long)256 * RR, 8, 128);
  for (int j = threadIdx.x; j < 8 * LD128; j += 256) vLds[8 * LD128 + j] = 0.f;
  async_wait0();
  __syncthreads();
  int w = threadIdx.x >> 5;
  int lane = threadIdx.x & 31, half = lane >> 4, col = lane & 15;
#pragma unroll
  for (int j = 0; j < 2; ++j) {
    int mt = w * 2 + j;
    v8f acc = {};
    for (int k0 = 0; k0 < 128; k0 += 4) {
      v2f a = frag_ks(zLds + mt * 16 * LD128 + k0, LD128, 1);
      v2f bb = frag_ks(vLds + k0, LD128, 1);
      acc = wmma4(a, bb, acc);
    }
    if (col < 8) {
      float* dst = rws + ((size_t)(b * KK + col) * 256 + (mt * 16 + 8 * half)) * 256 + n;
#pragma unroll
      for (int i = 0; i < 8; ++i) dst[(size_t)i * 256] += acc[i];
    }
  }
}

// ---- Kernel 5: fused  r += xt@yt^T ; softmax over m ; t = a^T @ xv ----------
__global__ void k_attn(const float* __restrict__ rws, const float* __restrict__ ws_xt,
                       const float* __restrict__ ws_yt, const float* __restrict__ ws_xv,
                       float* __restrict__ tws) {
  extern __shared__ float lds[];
  float* rbuf = lds;              // 256 x 32 score tile
  float* xbuf = lds + 256 * 32;   // 64 x LD64 (xt chunk, later xv chunk)
  float* ybuf = xbuf + 64 * LD64; // 32 x LD64 yt tile
  float* red = ybuf + 32 * LD64;  // 8 x 32
  float* cmax = red + 8 * 32;     // 32
  float* csum = cmax + 32;        // 32
  int blk = blockIdx.x;
  int b = blk >> 6;
  int k = (blk >> 3) & 7;
  int n0 = (blk & 7) * 32;
  int w = threadIdx.x >> 5;
  int t = threadIdx.x;

  cp_rows(ybuf, LD64, ws_yt + ((size_t)((b * KK + k) * 256 + n0)) * 64, 64, 32, 64);
  cp_rows(rbuf, 32, rws + ((size_t)(b * KK + k) * 256) * 256 + n0, 256, 256, 32);
  async_wait0();
  __syncthreads();

  // term1 accumulate: rbuf += xt @ yt^T
  int mt = w >> 1, ntl = w & 1;
  for (int mc = 0; mc < 4; ++mc) {
    cp_rows(xbuf, LD64, ws_xt + ((size_t)((b * KK + k) * 256 + mc * 64)) * 64, 64, 64, 64);
    async_wait0();
    __syncthreads();
    v8f acc = tile_load(rbuf + (mc * 64 + mt * 16) * 32 + ntl * 16, 32);
    for (int d0 = 0; d0 < 64; d0 += 4) {
      v2f a = frag_ks(xbuf + mt * 16 * LD64 + d0, LD64, 1);
      v2f bb = frag_ks(ybuf + ntl * 16 * LD64 + d0, LD64, 1);
      acc = wmma4(a, bb, acc);
    }
    tile_store(rbuf + (mc * 64 + mt * 16) * 32 + ntl * 16, 32, acc);
    __syncthreads();
  }

  // column softmax over m (TEMP = sqrt(64) = 8)
  {
    int col = t & 31, seg = t >> 5;
    float lm = -3.4e38f;
    for (int j = 0; j < 32; ++j) lm = fmaxf(lm, rbuf[(seg * 32 + j) * 32 + col]);
    red[seg * 32 + col] = lm;
    __syncthreads();
    if (t < 32) {
      float cm = -3.4e38f;
      for (int s = 0; s < 8; ++s) cm = fmaxf(cm, red[s * 32 + t]);
      cmax[t] = cm;
    }
    __syncthreads();
    const float invT = 0.125f;
    float cm = cmax[col];
    float ls = 0.f;
    for (int j = 0; j < 32; ++j) {
      float e = __expf((rbuf[(seg * 32 + j) * 32 + col] - cm) * invT);
      rbuf[(seg * 32 + j) * 32 + col] = e;
      ls += e;
    }
    red[seg * 32 + col] = ls;
    __syncthreads();
    if (t < 32) {
      float cs = 0.f;
      for (int s = 0; s < 8; ++s) cs += red[s * 32 + t];
      csum[t] = cs;
    }
    __syncthreads();
    float inv = 1.0f / csum[col];
    for (int j = 0; j < 32; ++j) rbuf[(seg * 32 + j) * 32 + col] *= inv;
  }
  __syncthreads();

  // t_tile[32 n x 64 d] = a^T(32x256) @ xv(256x64)
  int ntl2 = w >> 2, dt = w & 3;
  v8f acc = {};
  for (int mc = 0; mc < 4; ++mc) {
    __syncthreads();
    cp_rows(xbuf, LD64, ws_xv + ((size_t)((b * KK + k) * 256 + mc * 64)) * 64, 64, 64, 64);
    async_wait0();
    __syncthreads();
    for (int k0 = 0; k0 < 64; k0 += 4) {
      v2f a = frag_ks(rbuf + (mc * 64 + k0) * 32 + ntl2 * 16, 1, 32); // a^T access
      v2f bb = frag_ks(xbuf + k0 * LD64 + dt * 16, 1, LD64);          // xv transposed access
      acc = wmma4(a, bb, acc);
    }
  }
  tile_store(tws + ((size_t)((b * KK + k) * 256 + n0 + ntl2 * 16)) * 64 + dt * 16, 64, acc);
}

// ---- Kernel 6: y[b,n,q] = sum_{k,d} t[b,k,n,d] * Theta_y[k,q,d] -------------
__global__ void k_out(const float* __restrict__ tws, const float* __restrict__ Theta_y,
                      float* __restrict__ y) {
  extern __shared__ float lds[];
  float* aLds = lds;             // 64 x LD64 t chunk
  float* bLds = lds + 64 * LD64; // 64 x LD64 Theta_y chunk [q][d]
  int blk = blockIdx.x;
  int b = blk >> 5;
  int nb = (blk >> 3) & 3;
  int qb = blk & 7;
  int w = threadIdx.x >> 5;
  v8f acc0 = {}, acc1 = {};
  for (int k = 0; k < 8; ++k) {
    __syncthreads();
    cp_rows(aLds, LD64, tws + ((size_t)((b * KK + k) * 256 + nb * 64)) * 64, 64, 64, 64);
    cp_rows(bLds, LD64, Theta_y + ((size_t)k * 512 + qb * 64) * 64, 64, 64, 64);
    if (k < 7)
      __builtin_prefetch(tws + ((size_t)((b * KK + k + 1) * 256 + nb * 64)) * 64, 0, 3);
    async_wait0();
    __syncthreads();
#pragma unroll
    for (int j = 0; j < 2; ++j) {
      int ti = w + j * 8;
      int nt = ti >> 2, qt = ti & 3;
      v8f acc = j ? acc1 : acc0;
      for (int d0 = 0; d0 < 64; d0 += 4) {
        v2f a = frag_ks(aLds + nt * 16 * LD64 + d0, LD64, 1);
        v2f bb = frag_ks(bLds + qt * 16 * LD64 + d0, LD64, 1);
        acc = wmma4(a, bb, acc);
      }
      if (j) acc1 = acc; else acc0 = acc;
    }
  }
#pragma unroll
  for (int j = 0; j < 2; ++j) {
    int ti = w + j * 8;
    int nt = ti >> 2, qt = ti & 3;
    tile_store(y + ((size_t)(b * 256 + nb * 64 + nt * 16)) * 512 + qb * 64 + qt * 16, 512,
               j ? acc1 : acc0);
  }
}

// ---- launch -----------------------------------------------------------------
extern "C" void kernel_launch(void* const* d_in, const int* in_sizes, int n_in,
                              void* d_out, int out_size, void* d_ws, size_t ws_size,
                              hipStream_t stream) {
  (void)in_sizes; (void)n_in; (void)out_size; (void)ws_size;
  const float* z = (const float*)d_in[0];
  const float* y_q = (const float*)d_in[1];
  const float* x_k = (const float*)d_in[2];
  const float* Theta_x = (const float*)d_in[3];
  const float* Theta_y = (const float*)d_in[4];
  const float* Lam_x = (const float*)d_in[5];
  const float* Lam_y = (const float*)d_in[6];
  const float* Lam_zx = (const float*)d_in[7];
  const float* Lam_zy = (const float*)d_in[8];
  float* out = (float*)d_out;
  float* ws = (float*)d_ws;
  float* xt = ws + OFF_XT;
  float* yt = ws + OFF_YT;
  float* xv = ws + OFF_XV;
  float* u = ws + OFF_U;
  float* v = ws + OFF_V;
  float* rw = ws + OFF_R;
  float* tw = ws + OFF_T;

  dim3 blk(256, 1, 1);
  size_t lds_proj = (size_t)(64 * LD64 + 64 * 64) * sizeof(float);
  size_t lds_uv = (size_t)(64 * LD64 + 128 * LD64) * sizeof(float);
  size_t lds_r = (size_t)(256 * LD128 + 16 * LD128) * sizeof(float);
  size_t lds_attn = (size_t)(256 * 32 + 64 * LD64 + 32 * LD64 + 8 * 32 + 64) * sizeof(float);
  size_t lds_out = (size_t)(64 * LD64 + 64 * LD64) * sizeof(float);

  k_proj<<<dim3(192), blk, lds_proj, stream>>>(x_k, y_q, Lam_x, Lam_y, Theta_x, ws);
  k_uv<<<dim3(128), blk, lds_uv, stream>>>(xt, yt, Lam_zy, Lam_zx, u, v);
  k_r2<<<dim3(512), blk, lds_r, stream>>>(z, u, rw);
  k_r3<<<dim3(512), blk, lds_r, stream>>>(z, v, rw);
  k_attn<<<dim3(128), blk, lds_attn, stream>>>(rw, xt, yt, xv, tw);
  k_out<<<dim3(64), blk, lds_out, stream>>>(tw, Theta_y, out);
}